// SAGE_60258391163614
// MI455X (gfx1250) — compile-verified
//
#include <hip/hip_runtime.h>

// -------- problem constants (match reference) --------
#define N_NODES 50000
#define N_EDGES 800000
#define F_HIDDEN 128
#define F_OUTPUT 64

typedef __attribute__((ext_vector_type(2))) float v2f;
typedef __attribute__((ext_vector_type(8))) float v8f;

// ---------------------------------------------------------------------------
// zero workspace region (agg + cnt)
// ---------------------------------------------------------------------------
__global__ void zero_f32(float* __restrict__ p, int n) {
  int i = blockIdx.x * blockDim.x + threadIdx.x;
  if (i < n) p[i] = 0.0f;
}

// ---------------------------------------------------------------------------
// edge scatter: one wave32 per edge. 32 lanes x float4 = 128 floats/row.
// Gather of src row is a coalesced 512B read (L2-resident: node features are
// 25.6MB vs 192MB L2), accumulation via fp32 atomics resolved in L2.
// ---------------------------------------------------------------------------
__global__ __launch_bounds__(256) void scatter_add128(
    const float* __restrict__ x, const int* __restrict__ ei,
    float* __restrict__ agg, float* __restrict__ cnt, int E) {
  int t = blockIdx.x * blockDim.x + threadIdx.x;
  int e = t >> 5;
  if (e >= E) return;
  int lane = t & 31;
  int src = ei[e];       // ei row 0: source
  int dst = ei[E + e];   // ei row 1: destination
  const float4 v = ((const float4*)(x + (size_t)src * F_HIDDEN))[lane];
  float* ap = agg + (size_t)dst * F_HIDDEN + lane * 4;
  atomicAdd(ap + 0, v.x);
  atomicAdd(ap + 1, v.y);
  atomicAdd(ap + 2, v.z);
  atomicAdd(ap + 3, v.w);
  if (lane == 0) atomicAdd(cnt + dst, 1.0f);
}

// ---------------------------------------------------------------------------
// Fused SAGE GEMM: out = mean @ Wl + self @ Wr + b  (+relu, +residual)
// One block per 16-node tile; wave w computes output columns [16w, 16w+16).
// A' = [mean(16x128) | self(16x128)] staged in LDS, K=256 total.
// Matrix core: V_WMMA_F32_16X16X4_F32 (exact fp32, wave32).
//   A frag (16x4 f32): lane m = lane&15; v.x = A[m][kb+2*khalf],
//                      v.y = A[m][kb+2*khalf+1]   (khalf = lane>>4)
//   B frag (4x16 f32): mirrored rows, cols striped on lanes 0..15
//   C/D (16x16 f32):   elem r -> row r + 8*khalf, col lane&15
// Row stride 260 floats => stride mod 64 banks == 4 => fragment reads
// (16 rows x {k, k+2} columns) hit 32 distinct even banks: conflict-free.
// NOUT / RELU / HAS_RES are compile-time so all strides fold into immediate
// offsets (no v_mul_u64, no epilogue branches).
// ---------------------------------------------------------------------------
#define LDS_STRIDE 260

template <int NOUT, bool RELU, bool HAS_RES>
__global__ void sage_gemm(
    const float* __restrict__ agg, const float* __restrict__ cnt,
    const float* __restrict__ selfx,
    const float* __restrict__ Wl, const float* __restrict__ Wr,
    const float* __restrict__ bias, const float* __restrict__ residual,
    float* __restrict__ out) {
  __shared__ float Atile[16 * LDS_STRIDE];

  const int tid = threadIdx.x;
  const int nthreads = blockDim.x;
  const int mbase = blockIdx.x * 16;

  // Stage A' into LDS, coalesced; fuse mean = agg / max(cnt,1).
  for (int i = tid; i < 16 * 256; i += nthreads) {
    int r = i >> 8;
    int col = i & 255;
    int node = mbase + r;
    float v;
    if (col < F_HIDDEN) {
      float cv = cnt[node];
      v = agg[node * F_HIDDEN + col] * (1.0f / fmaxf(cv, 1.0f));
    } else {
      v = selfx[node * F_HIDDEN + (col - F_HIDDEN)];
    }
    Atile[r * LDS_STRIDE + col] = v;
  }
  __syncthreads();

  const int lane = tid & 31;
  const int wave = tid >> 5;
  const int khalf = lane >> 4;   // which K half-pair this lane holds
  const int nlane = lane & 15;   // output column within tile / A row index
  const int nbase = wave * 16;

  // accumulator seeded with bias (same column for all 8 rows of the frag)
  v8f c;
  {
    float bv = bias[nbase + nlane];
#pragma unroll
    for (int r = 0; r < 8; ++r) c[r] = bv;
  }

  const float* Arow = &Atile[nlane * LDS_STRIDE];

#pragma unroll 1
  for (int pass = 0; pass < 2; ++pass) {
    const float* __restrict__ W =
        (pass ? Wr : Wl) + nbase + nlane;          // [128 x NOUT] row-major
    const float* __restrict__ Acol = Arow + pass * F_HIDDEN;
#pragma unroll 8
    for (int kb = 0; kb < F_HIDDEN; kb += 4) {
      int k = kb + 2 * khalf;
      v2f a;
      a.x = Acol[k];
      a.y = Acol[k + 1];
      v2f b;
      b.x = W[k * NOUT];
      b.y = W[(k + 1) * NOUT];
      // 8 args: (neg_a, A, neg_b, B, c_mod, C, reuse_a, reuse_b)
      c = __builtin_amdgcn_wmma_f32_16x16x4_f32(false, a, false, b, (short)0, c,
                                                false, false);
    }
  }

  // epilogue: relu -> residual -> store (branch-free, immediate offsets)
  const int base = (mbase + 8 * khalf) * NOUT + nbase + nlane;
  float* __restrict__ op = out + base;
  const float* __restrict__ rp = HAS_RES ? residual + base : nullptr;
#pragma unroll
  for (int r = 0; r < 8; ++r) {
    float v = c[r];
    if (RELU) v = fmaxf(v, 0.0f);
    if (HAS_RES) v += rp[r * NOUT];
    op[r * NOUT] = v;
  }
}

// ---------------------------------------------------------------------------
extern "C" void kernel_launch(void* const* d_in, const int* in_sizes, int n_in,
                              void* d_out, int out_size, void* d_ws, size_t ws_size,
                              hipStream_t stream) {
  const float* x   = (const float*)d_in[0];
  const int*   ei1 = (const int*)d_in[1];
  const int*   ei2 = (const int*)d_in[2];
  const int*   ei3 = (const int*)d_in[3];
  const float* Wl1 = (const float*)d_in[4];
  const float* Wr1 = (const float*)d_in[5];
  const float* b1  = (const float*)d_in[6];
  const float* Wl2 = (const float*)d_in[7];
  const float* Wr2 = (const float*)d_in[8];
  const float* b2  = (const float*)d_in[9];
  const float* Wl3 = (const float*)d_in[10];
  const float* Wr3 = (const float*)d_in[11];
  const float* b3  = (const float*)d_in[12];

  // workspace layout (floats): agg[N*128] | cnt[N] | h1[N*128] | h2[N*128]
  float* agg = (float*)d_ws;
  float* cnt = agg + (size_t)N_NODES * F_HIDDEN;
  float* h1  = cnt + N_NODES;
  float* h2  = h1 + (size_t)N_NODES * F_HIDDEN;
  float* out = (float*)d_out;

  const int zn = N_NODES * F_HIDDEN + N_NODES;  // agg + cnt are contiguous
  dim3 zgrid((zn + 255) / 256);
  dim3 sgrid(((size_t)N_EDGES * 32 + 255) / 256);
  dim3 ggrid(N_NODES / 16);  // 50000 = 3125 * 16 exactly

  // ---- layer 1: h1 = relu(sage_conv(x, ei1)) ----
  zero_f32<<<zgrid, 256, 0, stream>>>(agg, zn);
  scatter_add128<<<sgrid, 256, 0, stream>>>(x, ei1, agg, cnt, N_EDGES);
  sage_gemm<F_HIDDEN, true, false>
      <<<ggrid, 256, 0, stream>>>(agg, cnt, x, Wl1, Wr1, b1, nullptr, h1);

  // ---- layer 2: h2 = relu(sage_conv(h1, ei2)) + h1 ----
  zero_f32<<<zgrid, 256, 0, stream>>>(agg, zn);
  scatter_add128<<<sgrid, 256, 0, stream>>>(h1, ei2, agg, cnt, N_EDGES);
  sage_gemm<F_HIDDEN, true, true>
      <<<ggrid, 256, 0, stream>>>(agg, cnt, h1, Wl2, Wr2, b2, h1, h2);

  // ---- layer 3: out = sage_conv(h2, ei3), F_OUT=64, no relu ----
  zero_f32<<<zgrid, 256, 0, stream>>>(agg, zn);
  scatter_add128<<<sgrid, 256, 0, stream>>>(h2, ei3, agg, cnt, N_EDGES);
  sage_gemm<F_OUTPUT, false, false>
      <<<ggrid, 128, 0, stream>>>(agg, cnt, h2, Wl3, Wr3, b3, nullptr, out);
}